// KMISPool_6640019440470
// MI455X (gfx1250) — compile-verified
//
#include <hip/hip_runtime.h>
#include <stdint.h>

// K-MIS pooling pipeline for MI455X (gfx1250, wave32).
// Stages: node rank (radix sort of scores) -> fixed-round MIS scatter-min/max ->
// cluster relabel (prefix sums) -> feature row gather via CDNA5 async LDS DMA ->
// edge lexsort (radix) -> segmented sum / coalesce -> outputs.
// CDNA5-specific paths:
//  * global_load_async_to_lds_b128 / global_store_async_from_lds_b128 + s_wait_asynccnt
//    for the 51MB feature-row gather (pure DMA permutation, HBM->LDS->HBM).
//  * v_wmma_f32_16x16x4_f32 tensor-core prefix scan: each 256-element block scan is
//    R = S(16x16) x U(upper-tri ones), chained over K in 4 WMMAs; counts < 2^24 so
//    f32 matrix math is exact. Used inside every radix pass + segment/compact scans.

typedef unsigned long long u64;
typedef unsigned int u32;
typedef float v2f __attribute__((ext_vector_type(2)));
typedef float v8f __attribute__((ext_vector_type(8)));

#define TPB 256
#define MIS_ROUNDS 24
#define CL_BITS 20  // cluster ids < 2^20 (n <= 1M); edge key = (r<<20)|c, 40 bits -> 5 radix passes

static __device__ __forceinline__ int digit_of(u64 k, int shift) {
  return (int)((k >> shift) & 255ull);
}

// ---------------- key construction ----------------

__global__ void kmis_node_keys(const float* __restrict__ score, u64* __restrict__ key,
                               u32* __restrict__ pay, int n) {
  int i = blockIdx.x * TPB + threadIdx.x;
  if (i < n) {
    u32 b = __float_as_uint(score[i]);
    u32 asc = (b & 0x80000000u) ? ~b : (b | 0x80000000u);  // ascending order map
    key[i] = (u64)(u32)(~asc);                              // descending score
    pay[i] = (u32)i;
  }
}

__global__ void kmis_edge_keys(const int* __restrict__ row, const int* __restrict__ col,
                               const int* __restrict__ cluster, u64* __restrict__ key,
                               u32* __restrict__ pay, int e) {
  int i = blockIdx.x * TPB + threadIdx.x;
  if (i < e) {
    u32 r2 = (u32)cluster[row[i]];
    u32 c2 = (u32)cluster[col[i]];
    key[i] = ((u64)r2 << CL_BITS) | (u64)c2;
    pay[i] = (u32)i;
  }
}

// ---------------- stable LSD radix sort (8-bit digits) ----------------

__global__ void kmis_hist(const u64* __restrict__ key, u32* __restrict__ hist,
                          int M, int shift, int B) {
  __shared__ u32 cnt[256];
  cnt[threadIdx.x] = 0;
  __syncthreads();
  int i = blockIdx.x * TPB + threadIdx.x;
  if (i < M) atomicAdd(&cnt[digit_of(key[i], shift)], 1u);
  __syncthreads();
  // column-major: hist[digit * B + block] so the scan yields stable global offsets
  hist[(size_t)threadIdx.x * (size_t)B + blockIdx.x] = cnt[threadIdx.x];
}

// WMMA tensor-core block scan: one wave32 per 256-element chunk.
// S[r][c] = d[base + r*16 + c]; R = S x U gives row-wise inclusive scan;
// 16 row totals get a tiny LDS scan for the cross-row offsets.
__global__ void kmis_scanA(u32* __restrict__ d, u32* __restrict__ part, int M) {
  __shared__ float rowtot[16];
  const int lane = (int)(threadIdx.x & 31u);
  const int m = lane & 15;   // A-row (M dim) == B/C/D column
  const int h = lane >> 4;   // half: selects K split (A) / row split (B,D)
  const size_t base = (size_t)blockIdx.x * 256;
#if defined(__gfx1250__) && defined(__HIP_DEVICE_COMPILE__)
  union { v8f v; float f[8]; } acc;
  acc.v = (v8f){};
  const size_t lim = (size_t)(M - 1);
  // K-chained WMMAs; all loads branchless (clamped addr + select) so EXEC stays
  // all-ones through the WMMA chain as the ISA requires.
  for (int t = 0; t < 4; ++t) {
    const int c0 = 4 * t + 2 * h;           // this lane's two K columns of S
    size_t i0 = base + (size_t)m * 16 + (size_t)c0;
    size_t i1 = i0 + 1;
    size_t j0 = i0 > lim ? lim : i0;
    size_t j1 = i1 > lim ? lim : i1;
    float x0 = (float)d[j0];
    float x1 = (float)d[j1];
    v2f a;
    a.x = (i0 <= lim) ? x0 : 0.f;
    a.y = (i1 <= lim) ? x1 : 0.f;
    // B fragment of U (4x16 tile, rows 4t..4t+3): VGPR v holds row v+2h, col m
    v2f b;
    b.x = ((4 * t + 2 * h + 0) <= m) ? 1.f : 0.f;
    b.y = ((4 * t + 2 * h + 1) <= m) ? 1.f : 0.f;
    acc.v = __builtin_amdgcn_wmma_f32_16x16x4_f32(
        /*neg_a=*/false, a, /*neg_b=*/false, b,
        /*c_mod=*/(short)0, acc.v, /*reuse_a=*/false, /*reuse_b=*/false);
  }
  // acc.f[v] = R[v + 8h][m]; column 15 holds the row-inclusive totals
  if (m == 15) {
    for (int v = 0; v < 8; ++v) rowtot[v + 8 * h] = acc.f[v];
  }
  __syncthreads();
  for (int v = 0; v < 8; ++v) {
    const int r = v + 8 * h;
    float off = 0.f;
    for (int k = 0; k < 16; ++k) off += (k < r) ? rowtot[k] : 0.f;
    size_t flat = base + (size_t)r * 16 + (size_t)m;
    if (flat < (size_t)M) d[flat] = (u32)(acc.f[v] + off + 0.5f);
  }
  if (lane == 0) {
    float tot = 0.f;
    for (int k = 0; k < 16; ++k) tot += rowtot[k];
    part[blockIdx.x] = (u32)(tot + 0.5f);
  }
#else
  // host-semantics / non-gfx1250 fallback: serial chunk scan
  if (lane == 0) {
    u32 run = 0;
    for (int k = 0; k < 256; ++k) {
      size_t idx = base + (size_t)k;
      if (idx < (size_t)M) { run += d[idx]; d[idx] = run; }
    }
    part[blockIdx.x] = run;
  }
  (void)rowtot; (void)m; (void)h;
#endif
}

__global__ void kmis_addback(u32* __restrict__ d, const u32* __restrict__ part, int M) {
  int i = blockIdx.x * TPB + threadIdx.x;
  if (blockIdx.x > 0 && i < M) d[i] += part[blockIdx.x - 1];
}

__global__ void kmis_scatter(const u64* __restrict__ keyIn, const u32* __restrict__ payIn,
                             u64* __restrict__ keyOut, u32* __restrict__ payOut,
                             const u32* __restrict__ histS, int M, int shift, int B) {
  __shared__ int dig[256];
  int i = blockIdx.x * TPB + threadIdx.x;
  int d = -1;
  u64 kv = 0;
  u32 pv = 0;
  if (i < M) {
    kv = keyIn[i];
    pv = payIn[i];
    d = digit_of(kv, shift);
  }
  dig[threadIdx.x] = d;
  __syncthreads();
  if (i < M) {
    int r = 0;
    for (int t = 0; t < (int)threadIdx.x; ++t) r += (dig[t] == d) ? 1 : 0;
    size_t idx0 = (size_t)d * (size_t)B + blockIdx.x;
    u32 base = (idx0 == 0) ? 0u : histS[idx0 - 1];  // exclusive from inclusive scan
    keyOut[base + r] = kv;
    payOut[base + r] = pv;
  }
}

// ---------------- MIS iteration ----------------

__global__ void kmis_rank(const u32* __restrict__ pay, int* __restrict__ rank, int n) {
  int i = blockIdx.x * TPB + threadIdx.x;
  if (i < n) rank[pay[i]] = i;
}

__global__ void kmis_init(const int* __restrict__ rank, int* __restrict__ mis,
                          int* __restrict__ minr, int n) {
  int i = blockIdx.x * TPB + threadIdx.x;
  if (i < n) {
    mis[i] = 0;
    minr[i] = rank[i];
  }
}

__global__ void kmis_copy_i32(const int* __restrict__ src, int* __restrict__ dst, int M) {
  int i = blockIdx.x * TPB + threadIdx.x;
  if (i < M) dst[i] = src[i];
}

__global__ void kmis_prop_min(const int* __restrict__ row, const int* __restrict__ col,
                              const int* __restrict__ oldv, int* __restrict__ cur, int e) {
  int i = blockIdx.x * TPB + threadIdx.x;
  if (i < e) atomicMin(&cur[col[i]], oldv[row[i]]);
}

__global__ void kmis_prop_max(const int* __restrict__ row, const int* __restrict__ col,
                              const int* __restrict__ oldv, int* __restrict__ cur, int e) {
  int i = blockIdx.x * TPB + threadIdx.x;
  if (i < e) atomicMax(&cur[col[i]], oldv[row[i]]);
}

__global__ void kmis_update1(const int* __restrict__ rank, const int* __restrict__ minr,
                             int* __restrict__ mis, int* __restrict__ mval, int n) {
  int i = blockIdx.x * TPB + threadIdx.x;
  if (i < n) {
    int m = mis[i] | (rank[i] == minr[i] ? 1 : 0);
    mis[i] = m;
    mval[i] = m;
  }
}

__global__ void kmis_update2(const int* __restrict__ rank, const int* __restrict__ mval,
                             int* __restrict__ minr, int n) {
  int i = blockIdx.x * TPB + threadIdx.x;
  if (i < n) minr[i] = (mval[i] > 0) ? n : rank[i];
}

// ---------------- clustering ----------------

__global__ void kmis_init_minc(const int* __restrict__ rank, const int* __restrict__ mis,
                               int* __restrict__ minc, int n) {
  int i = blockIdx.x * TPB + threadIdx.x;
  if (i < n) minc[i] = mis[i] ? rank[i] : n;
}

__global__ void kmis_fflag(const int* __restrict__ rank, const int* __restrict__ mis,
                           u32* __restrict__ f, int n) {
  int i = blockIdx.x * TPB + threadIdx.x;
  if (i < n) f[rank[i]] = (u32)mis[i];
}

__global__ void kmis_cluster(const int* __restrict__ minc, const u32* __restrict__ fincl,
                             int* __restrict__ cluster, int n) {
  int i = blockIdx.x * TPB + threadIdx.x;
  if (i < n) {
    int v = minc[i];
    cluster[i] = (v >= 0 && v < n) ? (int)fincl[v] - 1 : 0;  // searchsorted-left(rank)
  }
}

__global__ void kmis_mis_u32(const int* __restrict__ mis, u32* __restrict__ m2, int n) {
  int i = blockIdx.x * TPB + threadIdx.x;
  if (i < n) m2[i] = (u32)mis[i];
}

__global__ void kmis_compact(const int* __restrict__ mis, const u32* __restrict__ pos,
                             u32* __restrict__ idxArr, int n) {
  int i = blockIdx.x * TPB + threadIdx.x;
  if (i < n && mis[i]) idxArr[pos[i] - 1] = (u32)i;
}

// ---------------- feature gather via CDNA5 async LDS DMA ----------------

__global__ void kmis_gather_rows(const float* __restrict__ x, const u32* __restrict__ idxArr,
                                 const u32* __restrict__ misscan, float* __restrict__ out,
                                 int n, int f) {
  __shared__ __align__(16) float smem[8][128];  // one 512B row slot per wave
  const int c = (int)misscan[n - 1];
  const int lane = (int)(threadIdx.x & 31u);
  const int wave = (int)(threadIdx.x >> 5u);
  const int gw = (int)((blockIdx.x * blockDim.x + threadIdx.x) >> 5u);
  const int nw = (int)((gridDim.x * blockDim.x) >> 5u);
  for (int j = gw; j < n; j += nw) {
    float* dst = out + (size_t)j * (size_t)f;
    if (j < c) {
      const float* src = x + (size_t)idxArr[j] * (size_t)f;
      for (int base0 = 0; base0 < f; base0 += 128) {
        int base = base0 + lane * 4;
#if defined(__gfx1250__) && defined(__HIP_DEVICE_COMPILE__)
        typedef __attribute__((address_space(3))) float lds_f;
        lds_f* lp = (lds_f*)&smem[wave][lane * 4];
        u32 ldsa = (u32)(size_t)lp;                       // LDS byte offset
        u64 gin = (u64)(size_t)(src + base);              // global src addr
        u64 gout = (u64)(size_t)(dst + base);             // global dst addr
        // HBM -> LDS (ASYNCcnt tracked), then LDS -> HBM, no VGPR round trip
        asm volatile("global_load_async_to_lds_b128 %0, %1, off"
                     :: "v"(ldsa), "v"(gin) : "memory");
        asm volatile("s_wait_asynccnt 0x0" ::: "memory");
        asm volatile("global_store_async_from_lds_b128 %0, %1, off"
                     :: "v"(gout), "v"(ldsa) : "memory");
        asm volatile("s_wait_asynccnt 0x0" ::: "memory");  // LDS slot reuse hazard
#else
        float4 v = *(const float4*)(src + base);
        *(float4*)(dst + base) = v;
#endif
      }
    } else {
      float4 z = {0.f, 0.f, 0.f, 0.f};
      for (int base0 = 0; base0 < f; base0 += 128)
        *(float4*)(dst + base0 + lane * 4) = z;
    }
  }
}

// ---------------- pooled adjacency ----------------

__global__ void kmis_flags(const u64* __restrict__ keyS, u32* __restrict__ seg, int e) {
  int i = blockIdx.x * TPB + threadIdx.x;
  if (i < e) seg[i] = (i == 0 || keyS[i] != keyS[i - 1]) ? 1u : 0u;
}

__global__ void kmis_init_edge(float* __restrict__ outR, float* __restrict__ outC,
                               float* __restrict__ tempsum, int e) {
  int i = blockIdx.x * TPB + threadIdx.x;
  if (i < e) {
    outR[i] = -1.0f;
    outC[i] = -1.0f;
    tempsum[i] = 0.0f;
  }
}

__global__ void kmis_seg_heads(const u64* __restrict__ keyS, const u32* __restrict__ seg,
                               float* __restrict__ outR, float* __restrict__ outC, int e) {
  int i = blockIdx.x * TPB + threadIdx.x;
  if (i < e) {
    bool flag = (i == 0) || (keyS[i] != keyS[i - 1]);
    if (flag) {
      u32 s = seg[i] - 1;
      int r = (int)(keyS[i] >> CL_BITS);
      int cc = (int)(keyS[i] & ((1u << CL_BITS) - 1u));
      if (r != cc) {  // remove self loops
        outR[s] = (float)r;
        outC[s] = (float)cc;
      }
    }
  }
}

__global__ void kmis_seg_sum(const u32* __restrict__ payS, const float* __restrict__ attr,
                             const u32* __restrict__ seg, float* __restrict__ tempsum, int e) {
  int i = blockIdx.x * TPB + threadIdx.x;
  if (i < e) atomicAdd(&tempsum[seg[i] - 1], attr[payS[i]]);
}

__global__ void kmis_val_out(const float* __restrict__ outR, const float* __restrict__ tempsum,
                             float* __restrict__ val, int e) {
  int i = blockIdx.x * TPB + threadIdx.x;
  if (i < e) val[i] = (outR[i] >= 0.0f) ? tempsum[i] : 0.0f;
}

__global__ void kmis_mis_out(const int* __restrict__ mis, float* __restrict__ o, int n) {
  int i = blockIdx.x * TPB + threadIdx.x;
  if (i < n) o[i] = mis[i] ? 1.0f : 0.0f;
}

// ---------------- host driver ----------------

static void scan_u32(u32* d, u32* p1, u32* p2, u32* p3, int M, hipStream_t s) {
  int L1 = (M + TPB - 1) / TPB;
  kmis_scanA<<<L1, 32, 0, s>>>(d, p1, M);  // one wave32 per 256-chunk (WMMA scan)
  if (L1 > 1) {
    int L2 = (L1 + TPB - 1) / TPB;
    kmis_scanA<<<L2, 32, 0, s>>>(p1, p2, L1);
    if (L2 > 1) {
      kmis_scanA<<<1, 32, 0, s>>>(p2, p3, L2);  // L2 <= 256 for our sizes
      kmis_addback<<<L2, TPB, 0, s>>>(p1, p2, L1);
    }
    kmis_addback<<<L1, TPB, 0, s>>>(d, p1, M);
  }
}

static void radix_pass(const u64* kin, const u32* pin, u64* kout, u32* pout, u32* hist,
                       u32* p1, u32* p2, u32* p3, int M, int shift, hipStream_t s) {
  int B = (M + TPB - 1) / TPB;
  kmis_hist<<<B, TPB, 0, s>>>(kin, hist, M, shift, B);
  scan_u32(hist, p1, p2, p3, B * 256, s);
  kmis_scatter<<<B, TPB, 0, s>>>(kin, pin, kout, pout, hist, M, shift, B);
}

extern "C" void kernel_launch(void* const* d_in, const int* in_sizes, int n_in,
                              void* d_out, int out_size, void* d_ws, size_t ws_size,
                              hipStream_t stream) {
  const float* x = (const float*)d_in[0];
  const int* eidx = (const int*)d_in[1];
  const float* eattr = (const float*)d_in[2];
  const float* score = (const float*)d_in[3];
  const int n = in_sizes[3];
  const int e = in_sizes[2];
  const int f = in_sizes[0] / n;
  const int* row = eidx;
  const int* col = eidx + e;

  const int NB = (n + TPB - 1) / TPB;
  const int EB = (e + TPB - 1) / TPB;
  const int B_E = EB;

  // ---- workspace carve-out ----
  char* w = (char*)d_ws;
  size_t off = 0;
  auto alloc = [&](size_t bytes) -> void* {
    void* p = (void*)(w + off);
    off = (off + bytes + 255) & ~(size_t)255;
    return p;
  };
  u64* keyA = (u64*)alloc((size_t)e * 8);
  u64* keyB = (u64*)alloc((size_t)e * 8);
  u32* payA = (u32*)alloc((size_t)e * 4);
  u32* payB = (u32*)alloc((size_t)e * 4);
  u32* hist = (u32*)alloc((size_t)B_E * 256 * 4);
  u32* p1 = (u32*)alloc(16384 * 4);
  u32* p2 = (u32*)alloc(256 * 4);
  u32* p3 = (u32*)alloc(16 * 4);
  int* rank = (int*)alloc((size_t)n * 4);
  int* minr = (int*)alloc((size_t)n * 4);
  int* oldv = (int*)alloc((size_t)n * 4);
  int* mis = (int*)alloc((size_t)n * 4);
  int* mval = (int*)alloc((size_t)n * 4);
  int* minc = (int*)alloc((size_t)n * 4);
  int* cluster = (int*)alloc((size_t)n * 4);
  u32* fbuf = (u32*)alloc((size_t)n * 4);
  u32* m2 = (u32*)alloc((size_t)n * 4);
  u32* idxArr = (u32*)alloc((size_t)n * 4);

  // ---- output layout (flattened tuple, floats) ----
  float* o_x = (float*)d_out;                  // [n, f]
  float* o_r = o_x + (size_t)n * (size_t)f;    // [e]
  float* o_c = o_r + (size_t)e;                // [e]
  float* o_v = o_c + (size_t)e;                // [e]
  float* o_m = o_v + (size_t)e;                // [n]

  // ---- 1. rank = stable argsort(-score) positions (radix, 4 x 8-bit passes) ----
  kmis_node_keys<<<NB, TPB, 0, stream>>>(score, keyA, payA, n);
  radix_pass(keyA, payA, keyB, payB, hist, p1, p2, p3, n, 0, stream);
  radix_pass(keyB, payB, keyA, payA, hist, p1, p2, p3, n, 8, stream);
  radix_pass(keyA, payA, keyB, payB, hist, p1, p2, p3, n, 16, stream);
  radix_pass(keyB, payB, keyA, payA, hist, p1, p2, p3, n, 24, stream);  // final in A
  kmis_rank<<<NB, TPB, 0, stream>>>(payA, rank, n);

  // ---- 2. maximal 1-MIS: fixed rounds (no-op past convergence) ----
  kmis_init<<<NB, TPB, 0, stream>>>(rank, mis, minr, n);
  for (int it = 0; it < MIS_ROUNDS; ++it) {
    kmis_copy_i32<<<NB, TPB, 0, stream>>>(minr, oldv, n);
    kmis_prop_min<<<EB, TPB, 0, stream>>>(row, col, oldv, minr, e);
    kmis_update1<<<NB, TPB, 0, stream>>>(rank, minr, mis, mval, n);
    kmis_copy_i32<<<NB, TPB, 0, stream>>>(mval, oldv, n);
    kmis_prop_max<<<EB, TPB, 0, stream>>>(row, col, oldv, mval, e);
    kmis_update2<<<NB, TPB, 0, stream>>>(rank, mval, minr, n);
  }

  // ---- 3. cluster: nearest-MIS rank, relabel via prefix sum over rank space ----
  kmis_init_minc<<<NB, TPB, 0, stream>>>(rank, mis, minc, n);
  kmis_copy_i32<<<NB, TPB, 0, stream>>>(minc, oldv, n);
  kmis_prop_min<<<EB, TPB, 0, stream>>>(row, col, oldv, minc, e);
  kmis_fflag<<<NB, TPB, 0, stream>>>(rank, mis, fbuf, n);
  scan_u32(fbuf, p1, p2, p3, n, stream);
  kmis_cluster<<<NB, TPB, 0, stream>>>(minc, fbuf, cluster, n);

  // ---- 4. compact MIS node ids, gather features via async LDS DMA ----
  kmis_mis_u32<<<NB, TPB, 0, stream>>>(mis, m2, n);
  scan_u32(m2, p1, p2, p3, n, stream);
  kmis_compact<<<NB, TPB, 0, stream>>>(mis, m2, idxArr, n);
  kmis_gather_rows<<<1024, TPB, 0, stream>>>(x, idxArr, m2, o_x, n, f);

  // ---- 5. pooled adjacency: lexsort edges by (r2,c2), coalesce with sum ----
  kmis_edge_keys<<<EB, TPB, 0, stream>>>(row, col, cluster, keyA, payA, e);
  radix_pass(keyA, payA, keyB, payB, hist, p1, p2, p3, e, 0, stream);
  radix_pass(keyB, payB, keyA, payA, hist, p1, p2, p3, e, 8, stream);
  radix_pass(keyA, payA, keyB, payB, hist, p1, p2, p3, e, 16, stream);
  radix_pass(keyB, payB, keyA, payA, hist, p1, p2, p3, e, 24, stream);
  radix_pass(keyA, payA, keyB, payB, hist, p1, p2, p3, e, 32, stream);  // final in B
  u32* seg = (u32*)keyA;          // keyA/payA are free now: alias as seg / tempsum
  float* tsum = (float*)payA;
  kmis_flags<<<EB, TPB, 0, stream>>>(keyB, seg, e);
  scan_u32(seg, p1, p2, p3, e, stream);
  kmis_init_edge<<<EB, TPB, 0, stream>>>(o_r, o_c, tsum, e);
  kmis_seg_heads<<<EB, TPB, 0, stream>>>(keyB, seg, o_r, o_c, e);
  kmis_seg_sum<<<EB, TPB, 0, stream>>>(payB, eattr, seg, tsum, e);
  kmis_val_out<<<EB, TPB, 0, stream>>>(o_r, tsum, o_v, e);

  // ---- 6. mis mask ----
  kmis_mis_out<<<NB, TPB, 0, stream>>>(mis, o_m, n);
}